// PoolingLayer_24240795419245
// MI455X (gfx1250) — compile-verified
//
#include <hip/hip_runtime.h>

#define DFEAT 64
#define QPR   (DFEAT / 4)   // 16 float4 quads per feature row
#define TILE  256           // edges per block in scatter pass

// ---------------------------------------------------------------------------
// Hardware no-return f32 atomic add (device scope). __hip_atomic_fetch_add
// lowers to a CAS loop on this toolchain (verified via round-1 histogram:
// vmem count matched the CAS expansion exactly), so emit the CDNA5
// instruction directly: fire-and-forget, tracked by STOREcnt, RMW in L2.
// ---------------------------------------------------------------------------
__device__ __forceinline__ void atom_add_f32(float* p, float v) {
    unsigned long long a = (unsigned long long)(uintptr_t)p;
    asm volatile("global_atomic_add_f32 %0, %1, off scope:SCOPE_DEV"
                 :: "v"(a), "v"(v) : "memory");
}

__device__ __forceinline__ void atom_inc_u32(unsigned* p) {
    // integer atomics lower to global_atomic_add_u32 directly (verified)
    (void)__hip_atomic_fetch_add(p, 1u, __ATOMIC_RELAXED, __HIP_MEMORY_SCOPE_AGENT);
}

// ---------------------------------------------------------------------------
// Tensor Data Mover: DMA a 1-row tile of `tile_elems` u32 elements starting at
// gptr into LDS at byte offset lds_off. tensor_dim0 = rem_elems so reads past
// the end of the edge array return zero (ISA 8.2 OOB rule). Descriptor layout
// per CDNA5 ISA §8.3 (group0) / §8.4 (group1); groups 2/3 zero (<=2D tensor).
// ---------------------------------------------------------------------------
#if __has_builtin(__builtin_amdgcn_tensor_load_to_lds)
#define HAVE_TDM 1
typedef unsigned int u32x4 __attribute__((ext_vector_type(4)));
typedef int          i32x4 __attribute__((ext_vector_type(4)));
typedef int          i32x8 __attribute__((ext_vector_type(8)));

__device__ __forceinline__ void tdm_load_1d_u32(unsigned lds_off, const void* gptr,
                                                unsigned rem_elems, unsigned tile_elems) {
    unsigned long long ga = (unsigned long long)(uintptr_t)gptr;
    u32x4 g0;
    g0.x = 1u;                                     // count=1 valid descriptor
    g0.y = lds_off;                                // lds_addr (bytes)
    g0.z = (unsigned)(ga & 0xFFFFFFFFu);           // global_addr[31:0]
    g0.w = (unsigned)((ga >> 32) & 0x01FFFFFFu)    // global_addr[56:32]
         | (2u << 30);                             // type=2 ("image")
    i32x8 g1;
    g1[0] = (int)(2u << 16);                       // wg_mask=0, data_size=2 (4B)
    g1[1] = (int)((rem_elems & 0xFFFFu) << 16);    // tensor_dim0[15:0] @ [63:48]
    g1[2] = (int)((rem_elems >> 16) | (1u << 16)); // tensor_dim0[31:16] | tensor_dim1=1
    g1[3] = (int)((tile_elems & 0xFFFFu) << 16);   // tile_dim0 @ [127:112]
    g1[4] = 1;                                     // tile_dim1=1, tile_dim2=0
    g1[5] = (int)rem_elems;                        // tensor_dim0_stride[31:0]
    g1[6] = 0;                                     // stride hi / dim1_stride lo
    g1[7] = 0;
    i32x4 gz = {0, 0, 0, 0};
#if __has_include(<hip/amd_detail/amd_gfx1250_TDM.h>)
    i32x8 gz8 = {0, 0, 0, 0, 0, 0, 0, 0};          // amdgpu-toolchain: 6-arg form
    __builtin_amdgcn_tensor_load_to_lds(g0, g1, gz, gz, gz8, 0);
#else
    __builtin_amdgcn_tensor_load_to_lds(g0, g1, gz, gz, 0);  // ROCm 7.2: 5-arg form
#endif
}
#else
#define HAVE_TDM 0
#endif

// ---------------------------------------------------------------------------
// pass 2: gather + scatter-add. One block = TILE edges. Edge indices are
// staged into LDS once per block by the TDM (wave 0 issues two 1KB tile
// loads, waits on TENSORcnt, then the workgroup barrier publishes them);
// every thread then owns one 16B quad of an edge row per loop slot:
// global_load_b128 gather from xs (12.8MB, L2-resident) and 4 no-return
// global_atomic_add_f32 into the dst row.
// ---------------------------------------------------------------------------
__global__ void pm_scatter(const float* __restrict__ xs,
                           const int* __restrict__ src,
                           const int* __restrict__ dst,
                           float* __restrict__ out, int n_edges) {
    __shared__ int s_src[TILE];
    __shared__ int s_dst[TILE];

    const int tile_start = blockIdx.x * TILE;
    const unsigned rem = (unsigned)(n_edges - tile_start);   // > 0 by grid sizing

#if HAVE_TDM
    if (threadIdx.x < 32) {   // wave 0 only: TDM issue is per-wave, EXEC ignored
        tdm_load_1d_u32((unsigned)(uintptr_t)&s_src[0], src + tile_start, rem, TILE);
        tdm_load_1d_u32((unsigned)(uintptr_t)&s_dst[0], dst + tile_start, rem, TILE);
        __builtin_amdgcn_s_wait_tensorcnt(0);
    }
#else
    if (threadIdx.x < TILE) {
        int e = tile_start + threadIdx.x;
        s_src[threadIdx.x] = (e < n_edges) ? src[e] : 0;
        s_dst[threadIdx.x] = (e < n_edges) ? dst[e] : 0;
    }
#endif
    __syncthreads();

    const int slot = threadIdx.x >> 4;        // 0..15: edge lane-group
    const int q    = threadIdx.x & (QPR - 1); // quad within feature row

    #pragma unroll 4
    for (int j = 0; j < TILE / 16; ++j) {
        const int el = j * 16 + slot;
        const int e  = tile_start + el;
        if (e < n_edges) {
            const int s = s_src[el];
            const int d = s_dst[el];
            const float4 v = *reinterpret_cast<const float4*>(
                xs + (size_t)s * DFEAT + (size_t)q * 4);
            float* p = out + (size_t)d * DFEAT + (size_t)q * 4;
            atom_add_f32(p + 0, v.x);
            atom_add_f32(p + 1, v.y);
            atom_add_f32(p + 2, v.z);
            atom_add_f32(p + 3, v.w);
        }
    }
}

// ---- pass 0: zero output + counts ----------------------------------------
__global__ void pm_zero(float4* __restrict__ out4, unsigned* __restrict__ cnt,
                        int n_out4, int n_cnt) {
    int total = n_out4 + n_cnt;
    for (int i = blockIdx.x * blockDim.x + threadIdx.x; i < total;
         i += gridDim.x * blockDim.x) {
        if (i < n_out4) {
            out4[i] = make_float4(0.f, 0.f, 0.f, 0.f);
        } else {
            cnt[i - n_out4] = 0u;
        }
    }
}

// ---- pass 1: per-dst edge counts (u32 hardware atomics) -------------------
__global__ void pm_count(const int* __restrict__ dst, unsigned* __restrict__ cnt,
                         int n_edges) {
    int e = blockIdx.x * blockDim.x + threadIdx.x;
    if (e < n_edges) {
        atom_inc_u32(&cnt[dst[e]]);
    }
}

// ---- pass 3: divide by max(count, 1) --------------------------------------
__global__ void pm_divide(float4* __restrict__ out4,
                          const unsigned* __restrict__ cnt, int n_nodes) {
    int idx = blockIdx.x * blockDim.x + threadIdx.x;
    if (idx >= n_nodes * QPR) return;
    int node = idx >> 4;
    float c = (float)cnt[node];
    float sc = 1.0f / fmaxf(c, 1.0f);
    float4 v = out4[idx];
    v.x *= sc; v.y *= sc; v.z *= sc; v.w *= sc;
    out4[idx] = v;
}

// ---- launcher -------------------------------------------------------------
extern "C" void kernel_launch(void* const* d_in, const int* in_sizes, int n_in,
                              void* d_out, int out_size, void* d_ws, size_t ws_size,
                              hipStream_t stream) {
    const float* xs  = (const float*)d_in[0];
    const int*   src = (const int*)d_in[1];
    const int*   dst = (const int*)d_in[2];
    float*       out = (float*)d_out;

    const int n_nodes = in_sizes[0] / DFEAT;
    const int n_edges = in_sizes[1];

    unsigned* cnt = (unsigned*)d_ws;   // n_nodes * 4 bytes of scratch

    const int BS = 256;

    // pass 0: zero out + counts
    {
        int n_out4 = n_nodes * QPR;
        int total  = n_out4 + n_nodes;
        int grid   = (total + BS - 1) / BS;
        if (grid > 4096) grid = 4096;  // grid-stride
        pm_zero<<<grid, BS, 0, stream>>>((float4*)out, cnt, n_out4, n_nodes);
    }

    // pass 1: counts
    {
        int grid = (n_edges + BS - 1) / BS;
        pm_count<<<grid, BS, 0, stream>>>(dst, cnt, n_edges);
    }

    // pass 2: scatter-add (TILE edges per block)
    {
        int grid = (n_edges + TILE - 1) / TILE;
        pm_scatter<<<grid, BS, 0, stream>>>(xs, src, dst, out, n_edges);
    }

    // pass 3: divide
    {
        long long work = (long long)n_nodes * QPR;
        int grid = (int)((work + BS - 1) / BS);
        pm_divide<<<grid, BS, 0, stream>>>((float4*)out, cnt, n_nodes);
    }
}